// LCALModel_48404281426254
// MI455X (gfx1250) — compile-verified
//
#include <hip/hip_runtime.h>
#include <stdint.h>

#define MM 24
#define ZZ 1024
#define MZ (MM * ZZ)
#define P1_WAVES 4
#define P1_ROWS  8

typedef float v2f __attribute__((ext_vector_type(2)));
typedef float v8f __attribute__((ext_vector_type(8)));

__device__ __forceinline__ float slogf_(float x) { return __logf(fmaxf(x, 1e-12f)); }

// ---------------------------------------------------------------------------
// Precompute: ph, q = lamda*ph, c = slog(A) - beta*q, X_total, Xpred=0, loc[]
// ---------------------------------------------------------------------------
__global__ void k_pre(const float* __restrict__ h, const float* __restrict__ price,
                      const float* __restrict__ A, const float* __restrict__ beta,
                      const float* __restrict__ lamda, const float* __restrict__ X0,
                      const float* __restrict__ exogP,
                      const unsigned char* __restrict__ housing,
                      const unsigned char* __restrict__ genflux,
                      float* __restrict__ q, float* __restrict__ c,
                      float* __restrict__ Xt, float* __restrict__ Xpred,
                      int* __restrict__ loc)
{
    int idx = blockIdx.x * blockDim.x + threadIdx.x;
    if (idx >= MZ) return;
    int n = idx >> 10;
    float ph = price[idx] + h[idx];
    float qq = lamda[n] * ph;
    q[idx]  = qq;
    c[idx]  = slogf_(A[idx]) - beta[n] * qq;
    Xt[idx] = X0[idx] + exogP[idx];
    Xpred[idx] = 0.0f;
    if (idx < MM) loc[idx] = (genflux[idx] != 0 && housing[idx] == 0) ? 1 : 0;
}

// ---------------------------------------------------------------------------
// Eye sectors: U_ni = q[n,i] + t[n,i,i]  (only diagonal of t is touched)
// ---------------------------------------------------------------------------
__global__ void k_diag(const float* __restrict__ t, const float* __restrict__ q,
                       const int* __restrict__ loc, float* __restrict__ Uni)
{
    int idx = blockIdx.x * blockDim.x + threadIdx.x;
    if (idx >= MZ) return;
    int n = idx >> 10, i = idx & 1023;
    if (loc[n]) return;
    Uni[idx] = q[idx] + t[((size_t)idx << 10) + (size_t)i];
}

// ---------------------------------------------------------------------------
// Pass 1: per (n,i) row of t (4 KiB) streamed via async global->LDS copies,
// double-buffered per wave; softmax stats mx, sden, U_ni.
// ---------------------------------------------------------------------------
__global__ __launch_bounds__(128) void k_pass1(
    const float* __restrict__ t, const float* __restrict__ q,
    const float* __restrict__ c, const float* __restrict__ beta,
    const int* __restrict__ loc,
    float* __restrict__ mx, float* __restrict__ sden, float* __restrict__ Uni)
{
    __shared__ __align__(16) float smem[P1_WAVES][2][ZZ];
    const int n = blockIdx.x;
    if (!loc[n]) return;                    // eye sectors handled by k_diag
    const int wave = threadIdx.x >> 5;
    const int lane = threadIdx.x & 31;
    const float bn = beta[n];
    const float* crow = c + (n << 10);
    const float* qrow = q + (n << 10);
    const int ibase = (blockIdx.y * P1_WAVES + wave) * P1_ROWS;

    auto prefetch = [&](int i, int b) {
        const float* rowp = t + ((size_t)((n << 10) + i) << 10);
        uint32_t lbase = (uint32_t)(uintptr_t)(&smem[wave][b][0]);
        #pragma unroll
        for (int ch = 0; ch < 8; ++ch) {
            uint32_t laddr = lbase + (uint32_t)(ch * 512 + lane * 16);
            uint64_t gaddr = (uint64_t)(uintptr_t)rowp + (uint64_t)(ch * 512 + lane * 16);
            asm volatile("global_load_async_to_lds_b128 %0, %1, off"
                         :: "v"(laddr), "v"(gaddr) : "memory");
        }
    };

    prefetch(ibase, 0);
    for (int r = 0; r < P1_ROWS; ++r) {
        if (r + 1 < P1_ROWS) {
            prefetch(ibase + r + 1, (r + 1) & 1);
            asm volatile("s_wait_asynccnt 0x8" ::: "memory");   // current row's 8 done
        } else {
            asm volatile("s_wait_asynccnt 0x0" ::: "memory");
        }
        const float* buf = &smem[wave][r & 1][0];

        float lmax = -3.0e38f;
        #pragma unroll
        for (int k = 0; k < 8; ++k) {
            int j = lane * 4 + k * 128;
            float4 tv = *(const float4*)(buf + j);
            float4 cj = *(const float4*)(crow + j);
            lmax = fmaxf(lmax, fmaxf(fmaxf(cj.x - bn * tv.x, cj.y - bn * tv.y),
                                     fmaxf(cj.z - bn * tv.z, cj.w - bn * tv.w)));
        }
        #pragma unroll
        for (int o = 16; o >= 1; o >>= 1) lmax = fmaxf(lmax, __shfl_xor(lmax, o, 32));

        float se = 0.f, un = 0.f;
        #pragma unroll
        for (int k = 0; k < 8; ++k) {
            int j = lane * 4 + k * 128;
            float4 tv = *(const float4*)(buf + j);
            float4 cj = *(const float4*)(crow + j);
            float4 qj = *(const float4*)(qrow + j);
            float e;
            e = __expf(cj.x - bn * tv.x - lmax); se += e; un += e * (qj.x + tv.x);
            e = __expf(cj.y - bn * tv.y - lmax); se += e; un += e * (qj.y + tv.y);
            e = __expf(cj.z - bn * tv.z - lmax); se += e; un += e * (qj.z + tv.z);
            e = __expf(cj.w - bn * tv.w - lmax); se += e; un += e * (qj.w + tv.w);
        }
        #pragma unroll
        for (int o = 16; o >= 1; o >>= 1) {
            se += __shfl_xor(se, o, 32);
            un += __shfl_xor(un, o, 32);
        }
        if (lane == 0) {
            int row = (n << 10) + ibase + r;
            mx[row] = lmax; sden[row] = se; Uni[row] = un / se;
        }
    }
}

// ---------------------------------------------------------------------------
// Middle: per zone i, full 24x24 demand + masked substitution softmax -> D, v
// ---------------------------------------------------------------------------
__global__ void k_mid(const float* __restrict__ Uni, const float* __restrict__ Xt,
                      const float* __restrict__ attractor,
                      const float* __restrict__ demin, const float* __restrict__ demax,
                      const float* __restrict__ delta, const float* __restrict__ omega,
                      const float* __restrict__ sigma, const float* __restrict__ Kn,
                      const float* __restrict__ exogD, const float* __restrict__ sden,
                      float* __restrict__ Dni, float* __restrict__ vv)
{
    int i = blockIdx.x * blockDim.x + threadIdx.x;
    if (i >= ZZ) return;
    float Un[MM], sla[MM], Dacc[MM];
    #pragma unroll
    for (int n = 0; n < MM; ++n) {
        Un[n]  = Uni[(n << 10) + i];
        sla[n] = slogf_(attractor[(n << 10) + i]);
        Dacc[n] = 0.f;
    }
    for (int m = 0; m < MM; ++m) {
        float a_[MM], l_[MM];
        float sm = sigma[m];
        float lm = -3.0e38f;
        #pragma unroll
        for (int n = 0; n < MM; ++n) {
            int mn = m * MM + n;
            float dmn = demin[mn];
            float a = dmn + (demax[mn] - dmn) * __expf(-delta[mn] * Un[n]);
            a_[n] = a;
            float l = (Kn[mn] > 0.f) ? (sla[n] - sm * omega[mn] * a * Un[n]) : -1.0e30f;
            l_[n] = l;
            lm = fmaxf(lm, l);
        }
        float ssum = 0.f;
        #pragma unroll
        for (int n = 0; n < MM; ++n)
            if (l_[n] > -1.0e29f) ssum += __expf(l_[n] - lm);
        float xm = Xt[(m << 10) + i];
        #pragma unroll
        for (int n = 0; n < MM; ++n) {
            float S = (l_[n] > -1.0e29f) ? (__expf(l_[n] - lm) / ssum) : 1.0f;
            Dacc[n] += a_[n] * S * xm;
        }
    }
    #pragma unroll
    for (int n = 0; n < MM; ++n) {
        int idx = (n << 10) + i;
        float D = exogD[idx] + Dacc[n];
        Dni[idx] = D;
        vv[idx]  = D / sden[idx];       // only consumed for loc sectors
    }
}

// ---------------------------------------------------------------------------
// Pass 2: X_pred[n,j] = sum_i v[n,i]*exp(c[n,j] - beta*t[n,i,j] - mx[n,i])
// Rank-4 accumulation routed through v_wmma_f32_16x16x4_f32:
//   A(16x4) = w replicated over rows, B(4x16) = exp-terms (j striped on lanes)
// ---------------------------------------------------------------------------
__global__ __launch_bounds__(128) void k_pass2(
    const float* __restrict__ t, const float* __restrict__ c,
    const float* __restrict__ beta, const int* __restrict__ loc,
    const float* __restrict__ vv, const float* __restrict__ mx,
    float* __restrict__ Xpred)
{
    int bx = blockIdx.x;
    int n  = bx >> 6;          // 64 j-tiles of 16 per sector
    int jt = bx & 63;
    if (!loc[n]) return;
    int wave = threadIdx.x >> 5;
    int lane = threadIdx.x & 31;
    int half = lane >> 4;
    int j = (jt << 4) + (lane & 15);
    float cj = c[(n << 10) + j];
    float bn = beta[n];
    const float* vrow = vv + (n << 10);
    const float* mrow = mx + (n << 10);
    const float* trow = t + (((size_t)n) << 20);

    v8f acc = {0.f, 0.f, 0.f, 0.f, 0.f, 0.f, 0.f, 0.f};
    const int iend = (wave + 1) << 8;
    for (int i0 = wave << 8; i0 < iend; i0 += 4) {
        float4 v4 = *(const float4*)(vrow + i0);
        float4 m4 = *(const float4*)(mrow + i0);
        float va = half ? v4.z : v4.x;
        float vb = half ? v4.w : v4.y;
        float ma = half ? m4.z : m4.x;
        float mb = half ? m4.w : m4.y;
        int ia = i0 + (half << 1);
        float ta = trow[((size_t)ia << 10) + j];
        float tb = trow[((size_t)(ia + 1) << 10) + j];
        v2f A = { va, vb };
        v2f B = { __expf(cj - bn * ta - ma), __expf(cj - bn * tb - mb) };
        acc = __builtin_amdgcn_wmma_f32_16x16x4_f32(false, A, false, B,
                                                    (short)0, acc, false, false);
    }
    // D rows are replicated; row M=0 lives in VGPR0 of lanes 0..15 (N = lane)
    if (lane < 16) atomicAdd(&Xpred[(n << 10) + (jt << 4) + lane], acc[0]);
}

// ---------------------------------------------------------------------------
// MSE reduction
// ---------------------------------------------------------------------------
__global__ __launch_bounds__(1024) void k_mse(
    const float* __restrict__ Xpred, const float* __restrict__ Dni,
    const float* __restrict__ Xtar, const int* __restrict__ loc,
    float* __restrict__ out)
{
    __shared__ float red[1024];
    int tid = threadIdx.x;
    float s = 0.f;
    for (int idx = tid; idx < MZ; idx += 1024) {
        int n = idx >> 10;
        float xp = loc[n] ? Xpred[idx] : Dni[idx];   // eye sectors: X_pred = D
        float d = xp - Xtar[idx];
        s += d * d;
    }
    red[tid] = s; __syncthreads();
    for (int o = 512; o > 0; o >>= 1) {
        if (tid < o) red[tid] += red[tid + o];
        __syncthreads();
    }
    if (tid == 0) out[0] = red[0] / (float)MZ;
}

// ---------------------------------------------------------------------------
extern "C" void kernel_launch(void* const* d_in, const int* in_sizes, int n_in,
                              void* d_out, int out_size, void* d_ws, size_t ws_size,
                              hipStream_t stream)
{
    (void)in_sizes; (void)n_in; (void)out_size; (void)ws_size;
    const float* h      = (const float*)d_in[0];
    const float* price  = (const float*)d_in[1];
    const float* t      = (const float*)d_in[2];
    const float* demin  = (const float*)d_in[3];
    const float* demax  = (const float*)d_in[4];
    const float* delta  = (const float*)d_in[5];
    const float* omega  = (const float*)d_in[6];
    const float* sigma  = (const float*)d_in[7];
    const float* Kn     = (const float*)d_in[8];
    const float* attr   = (const float*)d_in[9];
    const float* beta   = (const float*)d_in[10];
    const float* lamda  = (const float*)d_in[11];
    const float* A_ni   = (const float*)d_in[12];
    const float* exogD  = (const float*)d_in[13];
    const float* exogP  = (const float*)d_in[14];
    const float* X0     = (const float*)d_in[15];
    const float* Xtar   = (const float*)d_in[16];
    const unsigned char* housing = (const unsigned char*)d_in[17];
    const unsigned char* genflux = (const unsigned char*)d_in[18];
    float* out = (float*)d_out;

    float* q     = (float*)d_ws;
    float* c     = q + MZ;
    float* Xt    = c + MZ;
    float* mxv   = Xt + MZ;
    float* sden  = mxv + MZ;
    float* Uni   = sden + MZ;
    float* Dni   = Uni + MZ;
    float* vv    = Dni + MZ;
    float* Xpred = vv + MZ;
    int*   loc   = (int*)(Xpred + MZ);

    k_pre <<<(MZ + 255) / 256, 256, 0, stream>>>(h, price, A_ni, beta, lamda, X0, exogP,
                                                 housing, genflux, q, c, Xt, Xpred, loc);
    k_diag<<<(MZ + 255) / 256, 256, 0, stream>>>(t, q, loc, Uni);
    k_pass1<<<dim3(MM, ZZ / (P1_WAVES * P1_ROWS)), 128, 0, stream>>>(t, q, c, beta, loc,
                                                                     mxv, sden, Uni);
    k_mid <<<ZZ / 256, 256, 0, stream>>>(Uni, Xt, attr, demin, demax, delta, omega,
                                         sigma, Kn, exogD, sden, Dni, vv);
    k_pass2<<<MM * 64, 128, 0, stream>>>(t, c, beta, loc, vv, mxv, Xpred);
    k_mse <<<1, 1024, 0, stream>>>(Xpred, Dni, Xtar, loc, out);
}